// TransitionDown_24988119728693
// MI455X (gfx1250) — compile-verified
//
#include <hip/hip_runtime.h>

#define B_    4
#define N_    8192
#define S_    2048
#define KNN_  16
#define CIN_  64
#define COUT_ 128
#define M_    (B_*S_*KNN_)   // 131072 rows through the MLP
#define EPS_  1e-5f

typedef __attribute__((ext_vector_type(2))) float v2f;
typedef __attribute__((ext_vector_type(8))) float v8f;

// ---------------------------------------------------------------------------
// Kernel 1: farthest point sampling. One block per batch; xyz + running
// distances live in registers (1024 thr * 8 pts). Writes new_xyz directly.
// ---------------------------------------------------------------------------
__global__ __launch_bounds__(1024) void fps_kernel(const float* __restrict__ xyz,
                                                   float* __restrict__ newxyz) {
  const int b = blockIdx.x;
  const int t = threadIdx.x;
  const int lane = t & 31, wv = t >> 5;
  __shared__ float cx, cy, cz;
  __shared__ float swv[32];
  __shared__ int   swi[32];
  __shared__ int   sfar;
  const float* xb = xyz + (size_t)b * N_ * 3;
  float px[8], py[8], pz[8], dist[8];
#pragma unroll
  for (int j = 0; j < 8; ++j) {
    int p = t + j * 1024;
    px[j] = xb[p*3+0]; py[j] = xb[p*3+1]; pz[j] = xb[p*3+2];
    dist[j] = 1e10f;
  }
  int far = 0;  // reference starts at index 0
  for (int s = 0; s < S_; ++s) {
    if ((far & 1023) == t) {          // owner broadcasts centroid coords
      int j = far >> 10;
      cx = px[j]; cy = py[j]; cz = pz[j];
    }
    __syncthreads();
    if (t < 3) {
      float v = (t == 0) ? cx : ((t == 1) ? cy : cz);
      newxyz[((size_t)b*S_ + s)*3 + t] = v;
    }
    float qx = cx, qy = cy, qz = cz;
    float bv = -1.f; int bi = 0;
#pragma unroll
    for (int j = 0; j < 8; ++j) {
      float dx = px[j]-qx, dy = py[j]-qy, dz = pz[j]-qz;
      float d = dx*dx + dy*dy + dz*dz;
      float nd = fminf(dist[j], d);
      dist[j] = nd;
      if (nd > bv) { bv = nd; bi = t + j*1024; }   // strict > keeps smallest idx
    }
#pragma unroll
    for (int off = 1; off < 32; off <<= 1) {       // wave32 argmax butterfly
      float ov = __shfl_xor(bv, off, 32);
      int   oi = __shfl_xor(bi, off, 32);
      if (ov > bv || (ov == bv && oi < bi)) { bv = ov; bi = oi; }
    }
    if (lane == 0) { swv[wv] = bv; swi[wv] = bi; }
    __syncthreads();
    if (wv == 0) {
      float v2 = swv[lane]; int i2 = swi[lane];
#pragma unroll
      for (int off = 1; off < 32; off <<= 1) {
        float ov = __shfl_xor(v2, off, 32);
        int   oi = __shfl_xor(i2, off, 32);
        if (ov > v2 || (ov == v2 && oi < i2)) { v2 = ov; i2 = oi; }
      }
      if (lane == 0) sfar = i2;
    }
    __syncthreads();
    far = sfar;
  }
}

// ---------------------------------------------------------------------------
// Kernel 2: exact KNN (K=16) per sampled point. One block per query; 32
// distances/thread in registers; 16 sequential argmin rounds == stable top-k.
// ---------------------------------------------------------------------------
__global__ __launch_bounds__(256) void knn_kernel(const float* __restrict__ xyz,
                                                  const float* __restrict__ newxyz,
                                                  int* __restrict__ knn) {
  const int q = blockIdx.x;                 // 0 .. B*S-1
  const int b = q >> 11;                    // q / 2048
  const int t = threadIdx.x;
  const int lane = t & 31, wv = t >> 5;
  __shared__ float swv[8];
  __shared__ int   swi[8];
  __shared__ int   swin;
  const float qx = newxyz[(size_t)q*3+0];
  const float qy = newxyz[(size_t)q*3+1];
  const float qz = newxyz[(size_t)q*3+2];
  const float* xb = xyz + (size_t)b * N_ * 3;
  float d[32];
#pragma unroll
  for (int i = 0; i < 32; ++i) {
    int p = t + i*256;
    float dx = xb[p*3+0]-qx, dy = xb[p*3+1]-qy, dz = xb[p*3+2]-qz;
    d[i] = dx*dx + dy*dy + dz*dz;
  }
  for (int r = 0; r < KNN_; ++r) {
    float bv = 3e38f; int bi = 0;
#pragma unroll
    for (int i = 0; i < 32; ++i)
      if (d[i] < bv) { bv = d[i]; bi = t + i*256; }
#pragma unroll
    for (int off = 1; off < 32; off <<= 1) {
      float ov = __shfl_xor(bv, off, 32);
      int   oi = __shfl_xor(bi, off, 32);
      if (ov < bv || (ov == bv && oi < bi)) { bv = ov; bi = oi; }
    }
    if (lane == 0) { swv[wv] = bv; swi[wv] = bi; }
    __syncthreads();
    if (wv == 0) {
      float v2 = (lane < 8) ? swv[lane] : 3e38f;
      int   i2 = (lane < 8) ? swi[lane] : 0x7fffffff;
#pragma unroll
      for (int off = 1; off < 8; off <<= 1) {
        float ov = __shfl_xor(v2, off, 32);
        int   oi = __shfl_xor(i2, off, 32);
        if (ov < v2 || (ov == v2 && oi < i2)) { v2 = ov; i2 = oi; }
      }
      if (lane == 0) { swin = i2; knn[(size_t)q*KNN_ + r] = i2; }
    }
    __syncthreads();
    int w = swin;
#pragma unroll
    for (int i = 0; i < 32; ++i)
      if (w == t + i*256) d[i] = 3e38f;     // exclude the winner
  }
}

// ---------------------------------------------------------------------------
// Kernel 3: fused gather + GEMM1 (X[16x68] @ W1[68x128], K padded 67->68)
// using V_WMMA_F32_16X16X4_F32. 8 waves each own a 16x16 tile of the 16x128
// block output. Also accumulates per-channel sum / sumsq for BatchNorm.
// ---------------------------------------------------------------------------
__global__ __launch_bounds__(256) void gemm1_kernel(const float* __restrict__ feat,
    const float* __restrict__ xyz, const float* __restrict__ newxyz,
    const int* __restrict__ knn, const float* __restrict__ W1,
    float* __restrict__ y1, float* __restrict__ stats) {
  __shared__ float As[68][17];              // column-major A tile, padded ld
  const int t = threadIdx.x;
  const int r = t >> 4, c0 = t & 15;
  const int m = blockIdx.x * 16 + r;
  const int s_ = (m >> 4) & (S_ - 1);
  const int b_ = m >> 15;
  const int nidx = knn[m];                  // knn is laid out exactly as rows of X
  const float* fp = feat + ((size_t)b_*N_ + nidx)*CIN_;
  As[c0     ][r] = fp[c0];
  As[c0 + 16][r] = fp[c0 + 16];
  As[c0 + 32][r] = fp[c0 + 32];
  As[c0 + 48][r] = fp[c0 + 48];
  if (c0 < 3)
    As[64 + c0][r] = xyz[((size_t)b_*N_ + nidx)*3 + c0]
                   - newxyz[((size_t)b_*S_ + s_)*3 + c0];
  if (c0 == 3) As[67][r] = 0.f;             // zero pad column K=67
  __syncthreads();

  const int lane = t & 31, wave = t >> 5;
  const int halo = lane >> 4, lr = lane & 15;
  const int colB = wave*16 + lr;
  v8f acc = {0.f,0.f,0.f,0.f,0.f,0.f,0.f,0.f};
#pragma unroll
  for (int kk = 0; kk < 68; kk += 4) {
    int k0 = kk + 2*halo;                   // lanes 0-15: K={kk,kk+1}; 16-31: {kk+2,kk+3}
    v2f a; a.x = As[k0][lr]; a.y = As[k0+1][lr];
    v2f bb;
    bb.x = (k0     < 67) ? W1[ k0     *COUT_ + colB] : 0.f;
    bb.y = (k0 + 1 < 67) ? W1[(k0 + 1)*COUT_ + colB] : 0.f;
    acc = __builtin_amdgcn_wmma_f32_16x16x4_f32(false, a, false, bb,
                                                (short)0, acc, false, false);
  }
  float sum = 0.f, sq = 0.f;
#pragma unroll
  for (int rr = 0; rr < 8; ++rr) {
    int row = rr + 8*halo;
    float v = acc[rr];
    y1[((size_t)(blockIdx.x*16 + row))*COUT_ + colB] = v;
    sum += v; sq += v*v;
  }
  sum += __shfl_xor(sum, 16, 32);
  sq  += __shfl_xor(sq,  16, 32);
  if (halo == 0) {
    atomicAdd(&stats[colB],        sum);
    atomicAdd(&stats[128 + colB],  sq);
  }
}

// ---------------------------------------------------------------------------
// Kernel 4: elementwise training-mode BatchNorm + ReLU (stats finalized here)
// ---------------------------------------------------------------------------
__global__ __launch_bounds__(256) void bn_relu_kernel(float* __restrict__ y,
    const float* __restrict__ stats, const float* __restrict__ g,
    const float* __restrict__ bta) {
  size_t e = (size_t)blockIdx.x*256 + threadIdx.x;
  int c = (int)(e & 127);
  float mu   = stats[c] * (1.f / M_);
  float var  = stats[128 + c] * (1.f / M_) - mu*mu;
  float rstd = rsqrtf(var + EPS_);
  float v = y[e];
  v = (v - mu) * rstd * g[c] + bta[c];
  y[e] = fmaxf(v, 0.f);
}

// ---------------------------------------------------------------------------
// Kernel 5: GEMM2 (Y1[16x128] @ W2[128x128]) with WMMA + stats accumulation
// ---------------------------------------------------------------------------
__global__ __launch_bounds__(256) void gemm2_kernel(const float* __restrict__ x,
    const float* __restrict__ W2, float* __restrict__ y2,
    float* __restrict__ stats) {
  __shared__ float As[128][17];
  const int t = threadIdx.x;
#pragma unroll
  for (int i = 0; i < 8; ++i) {
    int e = i*256 + t;
    int r = e >> 7, c = e & 127;
    As[c][r] = x[((size_t)(blockIdx.x*16 + r))*COUT_ + c];
  }
  __syncthreads();
  const int lane = t & 31, wave = t >> 5;
  const int halo = lane >> 4, lr = lane & 15;
  const int colB = wave*16 + lr;
  v8f acc = {0.f,0.f,0.f,0.f,0.f,0.f,0.f,0.f};
#pragma unroll
  for (int kk = 0; kk < 128; kk += 4) {
    int k0 = kk + 2*halo;
    v2f a; a.x = As[k0][lr]; a.y = As[k0+1][lr];
    v2f bb; bb.x = W2[k0*COUT_ + colB]; bb.y = W2[(k0+1)*COUT_ + colB];
    acc = __builtin_amdgcn_wmma_f32_16x16x4_f32(false, a, false, bb,
                                                (short)0, acc, false, false);
  }
  float sum = 0.f, sq = 0.f;
#pragma unroll
  for (int rr = 0; rr < 8; ++rr) {
    int row = rr + 8*halo;
    float v = acc[rr];
    y2[((size_t)(blockIdx.x*16 + row))*COUT_ + colB] = v;
    sum += v; sq += v*v;
  }
  sum += __shfl_xor(sum, 16, 32);
  sq  += __shfl_xor(sq,  16, 32);
  if (halo == 0) {
    atomicAdd(&stats[colB],       sum);
    atomicAdd(&stats[128 + colB], sq);
  }
}

// ---------------------------------------------------------------------------
// Kernel 6: BN2 + ReLU + max over K=16 -> new_feat
// ---------------------------------------------------------------------------
__global__ __launch_bounds__(256) void maxpool_kernel(const float* __restrict__ y2,
    const float* __restrict__ stats, const float* __restrict__ g,
    const float* __restrict__ bta, float* __restrict__ outfeat) {
  int e = blockIdx.x*256 + threadIdx.x;     // 0 .. B*S*128-1
  int c  = e & 127;
  int bs = e >> 7;
  float mu   = stats[c] * (1.f / M_);
  float var  = stats[128 + c] * (1.f / M_) - mu*mu;
  float rstd = rsqrtf(var + EPS_);
  float gg = g[c], bb = bta[c];
  float mx = 0.f;                            // relu outputs are >= 0
#pragma unroll
  for (int k = 0; k < KNN_; ++k) {
    float v = y2[((size_t)bs*KNN_ + k)*COUT_ + c];
    float r = fmaxf((v - mu)*rstd*gg + bb, 0.f);
    mx = fmaxf(mx, r);
  }
  outfeat[e] = mx;
}

// ---------------------------------------------------------------------------
extern "C" void kernel_launch(void* const* d_in, const int* in_sizes, int n_in,
                              void* d_out, int out_size, void* d_ws, size_t ws_size,
                              hipStream_t stream) {
  (void)in_sizes; (void)n_in; (void)out_size; (void)ws_size;
  const float* xyz  = (const float*)d_in[0];
  const float* feat = (const float*)d_in[1];
  const float* W1   = (const float*)d_in[2];
  const float* g1   = (const float*)d_in[3];
  const float* b1   = (const float*)d_in[4];
  const float* W2   = (const float*)d_in[5];
  const float* g2   = (const float*)d_in[6];
  const float* b2   = (const float*)d_in[7];

  float* out     = (float*)d_out;
  float* newxyz  = out;                     // [B,S,3]   -> 24576 floats
  float* newfeat = out + B_*S_*3;           // [B,S,128]

  char*  ws    = (char*)d_ws;
  float* stats = (float*)ws;                          // 512 f32: sum1,sq1,sum2,sq2
  int*   knn   = (int*)(ws + 4096);                   // [B,S,16]
  float* y1    = (float*)(ws + (1u << 20));           // [M,128]  64 MB
  float* y2    = (float*)(ws + (1u << 20) + ((size_t)64 << 20));  // [M,128] 64 MB

  hipMemsetAsync(stats, 0, 512*sizeof(float), stream);

  fps_kernel   <<<B_,            1024, 0, stream>>>(xyz, newxyz);
  knn_kernel   <<<B_*S_,          256, 0, stream>>>(xyz, newxyz, knn);
  gemm1_kernel <<<M_/16,          256, 0, stream>>>(feat, xyz, newxyz, knn, W1, y1, stats);
  bn_relu_kernel<<<(M_*COUT_)/256,256, 0, stream>>>(y1, stats, g1, b1);
  gemm2_kernel <<<M_/16,          256, 0, stream>>>(y1, W2, y2, stats + 256);
  maxpool_kernel<<<(B_*S_*COUT_)/256, 256, 0, stream>>>(y2, stats + 256, g2, b2, newfeat);
}